// SpectralFreqTimeConv2D_7172595384341
// MI455X (gfx1250) — compile-verified
//
#include <hip/hip_runtime.h>
#include <hip/hip_bf16.h>
#include <math.h>

typedef float v2f __attribute__((ext_vector_type(2)));
typedef float v8f __attribute__((ext_vector_type(8)));

// ---------------- problem constants ----------------
constexpr int BATCH = 32;
constexpr int NG    = 128;           // spatial grid
constexpr int CIN   = 32;
constexpr int COUT  = 32;
constexpr int TDIM  = 256;
constexpr int MODES = 17;            // kept modes per axis (N/2+1 truncated)
constexpr int P2    = 34;            // 2*MODES kept row-modes (0..16, 111..127)
constexpr int PM    = 48;            // P2 padded to 16-multiple
constexpr int NA    = NG * CIN;      // 4096: fused (v,c) N-dim for stage A
constexpr int QPAD  = 32;            // q padded
constexpr int KDP   = 36;            // stage-D K (p) padded to 4-multiple
constexpr int KEP   = 20;            // stage-E K (q) padded to 4-multiple
constexpr int NUQ   = MODES * COUT;  // 544: fused (q,o) N-dim for stage D

// LDS row pads (even, co-prime-ish with 64 banks to kill row-parallel conflicts)
constexpr int LPA = 130;             // stage A/B tables (K=128 cols)
constexpr int LPD = 38;              // stage D table (36 cols)
constexpr int LPE = 22;              // stage E tables (20 cols)

// ---------------- workspace layout (floats) ----------------
constexpr size_t OFF_F1R = 0;
constexpr size_t OFF_F1I = OFF_F1R + (size_t)PM * NG;
constexpr size_t OFF_F2R = OFF_F1I + (size_t)PM * NG;
constexpr size_t OFF_F2I = OFF_F2R + (size_t)QPAD * NG;
constexpr size_t OFF_G1R = OFF_F2I + (size_t)QPAD * NG;
constexpr size_t OFF_G1I = OFF_G1R + (size_t)NG * KDP;
constexpr size_t OFF_C2  = OFF_G1I + (size_t)NG * KDP;
constexpr size_t OFF_S2  = OFF_C2  + (size_t)NG * KEP;
constexpr size_t OFF_T   = OFF_S2  + (size_t)NG * KEP;     // t1r,t1i,t2r,t2i (4*544)
constexpr size_t OFF_Y   = ((OFF_T + 4 * (size_t)BATCH * MODES) + 127) & ~((size_t)127);
constexpr size_t YSZ     = (size_t)BATCH * P2 * NA;        // one complex plane
constexpr size_t OFF_Z   = OFF_Y + 2 * YSZ;
constexpr size_t ZSZ     = (size_t)BATCH * P2 * MODES * CIN;
constexpr size_t OFF_U   = OFF_Z + 2 * ZSZ;
constexpr size_t USZ     = (size_t)BATCH * P2 * MODES * COUT;
constexpr size_t VSZ     = (size_t)BATCH * NG * NUQ;       // overlaid on Y region
constexpr size_t WS_FLOATS = OFF_U + 2 * USZ;              // ~11.3M floats (~45 MB)

// ---------------- WMMA f32 16x16x4 wrapper (codegen-verified round 1) ------
static __device__ __forceinline__ v8f wmma4(v2f a, v2f b, v8f c) {
#if __has_builtin(__builtin_amdgcn_wmma_f32_16x16x4_f32)
  return __builtin_amdgcn_wmma_f32_16x16x4_f32(false, a, false, b, (short)0, c,
                                               false, false);
#else
  v8f d;
  asm volatile("v_wmma_f32_16x16x4_f32 %0, %1, %2, %3"
               : "=v"(d)
               : "v"(a), "v"(b), "v"(c));
  return d;
#endif
}

// ---------------- twiddle / basis tables ----------------
__global__ __launch_bounds__(256) void k_twiddle(float* __restrict__ ws) {
  const int i = blockIdx.x * blockDim.x + threadIdx.x;
  const float W = 6.28318530717958647692f / (float)NG;
  // F1: forward row-DFT, 34 kept modes (rows 0..16 and 111..127), 1/N^2 folded.
  if (i < PM * NG) {
    const int p = i / NG, u = i % NG;
    float vr = 0.f, vi = 0.f;
    if (p < P2) {
      const int r  = (p < MODES) ? p : p + (NG - 2 * MODES);  // p>=17 -> 111..127
      const float th = -W * (float)((r * u) & (NG - 1));
      const float s  = 1.0f / (float)(NG * NG);               // norm='forward'
      vr = s * cosf(th);
      vi = s * sinf(th);
    }
    ws[OFF_F1R + i] = vr;
    ws[OFF_F1I + i] = vi;
  }
  // F2: forward col-DFT, 17 kept modes (A matrix is [q][v]).
  if (i < QPAD * NG) {
    const int q = i / NG, v = i % NG;
    float vr = 0.f, vi = 0.f;
    if (q < MODES) {
      const float th = -W * (float)((q * v) & (NG - 1));
      vr = cosf(th);
      vi = sinf(th);
    }
    ws[OFF_F2R + i] = vr;
    ws[OFF_F2I + i] = vi;
  }
  // G1: inverse row-DFT [u][p], p over the 34 kept modes.
  if (i < NG * KDP) {
    const int u = i / KDP, p = i % KDP;
    float vr = 0.f, vi = 0.f;
    if (p < P2) {
      const int r  = (p < MODES) ? p : p + (NG - 2 * MODES);
      const float th = W * (float)((r * u) & (NG - 1));
      vr = cosf(th);
      vi = sinf(th);
    }
    ws[OFF_G1R + i] = vr;
    ws[OFF_G1I + i] = vi;
  }
  // C2/S2: inverse rfft over cols -> real; c_q weights folded (q=0: 1, else 2).
  if (i < NG * KEP) {
    const int v = i / KEP, q = i % KEP;
    float c = 0.f, s = 0.f;
    if (q < MODES) {
      const float cq = (q == 0) ? 1.f : 2.f;
      const float th = W * (float)((q * v) & (NG - 1));
      c = cq * cosf(th);
      s = -cq * sinf(th);
    }
    ws[OFF_C2 + i] = c;
    ws[OFF_S2 + i] = s;
  }
}

// ---------------- time-modulation coefficients t = t_emb @ k ----------------
__global__ __launch_bounds__(256) void k_tmod(const float* __restrict__ temb,
                                              const float* __restrict__ k1r,
                                              const float* __restrict__ k1i,
                                              const float* __restrict__ k2r,
                                              const float* __restrict__ k2i,
                                              float* __restrict__ ws) {
  const int i = blockIdx.x * blockDim.x + threadIdx.x;
  if (i >= 4 * BATCH * MODES) return;
  const int which = i / (BATCH * MODES);
  const int rem   = i % (BATCH * MODES);
  const int b = rem / MODES, m = rem % MODES;
  const float* k = (which == 0) ? k1r : (which == 1) ? k1i : (which == 2) ? k2r : k2i;
  float acc = 0.f;
  for (int t = 0; t < TDIM; ++t) acc += temb[b * TDIM + t] * k[t * MODES + m];
  ws[OFF_T + i] = acc;  // [which][b][m], contiguous
}

// ---------------- Stage A: Y[b,p,(v,c)] = F1 (cplx) x X (real) ----------------
// All 8 waves of a block share one 16-row A tile -> stage it in LDS (padded).
__global__ __launch_bounds__(256) void k_fwd_rows(const float* __restrict__ x,
                                                  float* __restrict__ ws) {
  __shared__ float lAr[16 * LPA];
  __shared__ float lAi[16 * LPA];

  const int bw  = (int)blockIdx.x * 8;        // first wave id of this block
  const int tb  = bw >> 8;                    // block-uniform (8 | 256)
  const int mtb = tb % 3;
  const int b   = tb / 3;
  const int m0  = mtb * 16;

  const float* F1r = ws + OFF_F1R;
  const float* F1i = ws + OFF_F1I;
  // Cooperative, coalesced stage of the shared A tile (2048 = 8*256 elems).
  for (int idx = threadIdx.x; idx < 16 * NG; idx += 256) {
    const int row = idx >> 7, col = idx & (NG - 1);
    lAr[row * LPA + col] = F1r[(m0 + row) * NG + col];
    lAi[row * LPA + col] = F1i[(m0 + row) * NG + col];
  }
  __syncthreads();

  const int wid  = bw + (int)(threadIdx.x >> 5);
  const int lane = (int)(threadIdx.x & 31);
  const int half = lane >> 4, l15 = lane & 15;
  const int nt = wid & 255;

  const float* X = x + (size_t)b * NG * NG * CIN;
  float* Yr = ws + OFF_Y;
  float* Yi = Yr + YSZ;

  const int n0 = nt * 16;
  const int bn = n0 + l15;  // B col (v*CIN+c)

  v8f dr = {}, di = {};
  for (int k0 = 0; k0 < NG; k0 += 4) {
    const int kb = k0 + half * 2;
    v2f ar = {lAr[l15 * LPA + kb], lAr[l15 * LPA + kb + 1]};
    v2f ai = {lAi[l15 * LPA + kb], lAi[l15 * LPA + kb + 1]};
    const float* bp = X + (size_t)kb * NA + bn;
    if (kb + 16 < NG) __builtin_prefetch(bp + (size_t)16 * NA, 0, 0);
    v2f bv = {bp[0], bp[NA]};
    dr = wmma4(ar, bv, dr);
    di = wmma4(ai, bv, di);
  }
  for (int r = 0; r < 8; ++r) {
    const int m = m0 + r + half * 8;
    if (m < P2) {
      const size_t idx = ((size_t)b * P2 + m) * NA + bn;
      Yr[idx] = dr[r];
      Yi[idx] = di[r];
    }
  }
}

// ---------------- Stage B: Z[b,p,q,c] = F2 (cplx) x Y (cplx) ----------------
// Whole 32x128 F2 table staged in LDS (shared by all waves / both q-tiles).
__global__ __launch_bounds__(256) void k_fwd_cols(float* __restrict__ ws) {
  __shared__ float lAr[QPAD * LPA];
  __shared__ float lAi[QPAD * LPA];

  const float* F2r = ws + OFF_F2R;
  const float* F2i = ws + OFF_F2I;
  for (int idx = threadIdx.x; idx < QPAD * NG; idx += 256) {  // 4096 = 16*256
    const int row = idx >> 7, col = idx & (NG - 1);
    lAr[row * LPA + col] = F2r[row * NG + col];
    lAi[row * LPA + col] = F2i[row * NG + col];
  }
  __syncthreads();

  const int wid  = (int)((blockIdx.x * blockDim.x + threadIdx.x) >> 5);
  const int lane = (int)(threadIdx.x & 31);
  const int half = lane >> 4, l15 = lane & 15;

  const int nt = wid & 1;
  const int mt = (wid >> 1) & 1;
  const int t  = wid >> 2;
  const int p  = t % P2;
  const int b  = t / P2;

  const float* Yr = ws + OFF_Y;
  const float* Yi = Yr + YSZ;
  float* Zr = ws + OFF_Z;
  float* Zi = Zr + ZSZ;

  const size_t ybase = ((size_t)b * P2 + p) * NA;
  const int m0 = mt * 16, n0 = nt * 16;
  const int am = m0 + l15;  // q
  const int bn = n0 + l15;  // c

  v8f dr = {}, di = {};
  for (int k0 = 0; k0 < NG; k0 += 4) {
    const int kb = k0 + half * 2;
    v2f ar = {lAr[am * LPA + kb], lAr[am * LPA + kb + 1]};
    v2f ai = {lAi[am * LPA + kb], lAi[am * LPA + kb + 1]};
    const float* brp = Yr + ybase + (size_t)kb * CIN + bn;
    const float* bip = Yi + ybase + (size_t)kb * CIN + bn;
    v2f br = {brp[0], brp[CIN]};
    v2f bi = {bip[0], bip[CIN]};
    v2f nai = -ai;
    dr = wmma4(ar, br, dr);
    dr = wmma4(nai, bi, dr);
    di = wmma4(ar, bi, di);
    di = wmma4(ai, br, di);
  }
  for (int r = 0; r < 8; ++r) {
    const int q = m0 + r + half * 8;
    if (q < MODES) {
      const size_t idx = (((size_t)b * P2 + p) * MODES + q) * CIN + bn;
      Zr[idx] = dr[r];
      Zi[idx] = di[r];
    }
  }
}

// ------- Stage C: U[b,p,q,o] = t[b,i(p)] * (Z[b,p,q,:] x W{1,2}[i,q,:,:]) -------
__global__ __launch_bounds__(256) void k_specmul(const float* __restrict__ w1r,
                                                 const float* __restrict__ w1i,
                                                 const float* __restrict__ w2r,
                                                 const float* __restrict__ w2i,
                                                 float* __restrict__ ws) {
  const int wid  = (int)((blockIdx.x * blockDim.x + threadIdx.x) >> 5);
  const int lane = (int)(threadIdx.x & 31);
  const int half = lane >> 4, l15 = lane & 15;

  const int nt = wid & 1;
  const int mt = (wid >> 1) & 1;
  const int t  = wid >> 2;
  const int q  = t % MODES;
  const int p  = t / MODES;

  const int idm = (p < MODES) ? p : p - MODES;  // weight mode index i
  const float* wr = (p < MODES) ? w1r : w2r;
  const float* wi = (p < MODES) ? w1i : w2i;
  const size_t wbase = ((size_t)idm * MODES + q) * CIN * COUT;

  const float* Zr = ws + OFF_Z;
  const float* Zi = Zr + ZSZ;
  float* Ur = ws + OFF_U;
  float* Ui = Ur + USZ;

  const int m0 = mt * 16, n0 = nt * 16;
  const int am = m0 + l15;  // batch row
  const int bn = n0 + l15;  // o
  const size_t zrow = (((size_t)am * P2 + p) * MODES + q) * CIN;

  v8f dr = {}, di = {};
  for (int k0 = 0; k0 < CIN; k0 += 4) {
    const int kb = k0 + half * 2;
    v2f ar = {Zr[zrow + kb], Zr[zrow + kb + 1]};
    v2f ai = {Zi[zrow + kb], Zi[zrow + kb + 1]};
    const float* brp = wr + wbase + (size_t)kb * COUT + bn;
    const float* bip = wi + wbase + (size_t)kb * COUT + bn;
    v2f br = {brp[0], brp[COUT]};
    v2f bi = {bip[0], bip[COUT]};
    v2f nai = -ai;
    dr = wmma4(ar, br, dr);
    dr = wmma4(nai, bi, dr);
    di = wmma4(ar, bi, di);
    di = wmma4(ai, br, di);
  }
  const float* tr_tab = (p < MODES) ? (ws + OFF_T) : (ws + OFF_T + 2 * BATCH * MODES);
  const float* ti_tab = (p < MODES) ? (ws + OFF_T + BATCH * MODES)
                                    : (ws + OFF_T + 3 * BATCH * MODES);
  for (int r = 0; r < 8; ++r) {
    const int bb = m0 + r + half * 8;
    const float tr = tr_tab[bb * MODES + idm];
    const float ti = ti_tab[bb * MODES + idm];
    const float urv = dr[r] * tr - di[r] * ti;
    const float uiv = dr[r] * ti + di[r] * tr;
    const size_t idx = (((size_t)bb * P2 + p) * MODES + q) * COUT + bn;
    Ur[idx] = urv;
    Ui[idx] = uiv;
  }
}

// ---------------- Stage D: V[b,u,(q,o)] = G1 (cplx) x U (cplx) ----------------
// Whole 128x36 G1 table staged in LDS (rows needed differ per wave).
__global__ __launch_bounds__(256) void k_inv_rows(float* __restrict__ ws) {
  __shared__ float lGr[NG * LPD];
  __shared__ float lGi[NG * LPD];

  const float* G1r = ws + OFF_G1R;
  const float* G1i = ws + OFF_G1I;
  for (int idx = threadIdx.x; idx < NG * KDP; idx += 256) {  // 4608 = 18*256
    const int row = idx / KDP, col = idx % KDP;
    lGr[row * LPD + col] = G1r[idx];
    lGi[row * LPD + col] = G1i[idx];
  }
  __syncthreads();

  const int wid  = (int)((blockIdx.x * blockDim.x + threadIdx.x) >> 5);
  const int lane = (int)(threadIdx.x & 31);
  const int half = lane >> 4, l15 = lane & 15;

  const int nt = wid % 34;
  const int t  = wid / 34;
  const int mt = t & 7;
  const int b  = t >> 3;

  const float* Ur = ws + OFF_U;
  const float* Ui = Ur + USZ;
  float* Vr = ws + OFF_Y;  // Y region is dead now: overlay
  float* Vi = Vr + VSZ;

  const int m0 = mt * 16, n0 = nt * 16;
  const int am = m0 + l15;  // u
  const int bn = n0 + l15;  // q*COUT+o

  v8f dr = {}, di = {};
  for (int k0 = 0; k0 < KDP; k0 += 4) {
    const int kb = k0 + half * 2;
    v2f ar = {lGr[am * LPD + kb], lGr[am * LPD + kb + 1]};
    v2f ai = {lGi[am * LPD + kb], lGi[am * LPD + kb + 1]};
    v2f br = {}, bi = {};
    if (kb < P2) {
      br.x = Ur[((size_t)b * P2 + kb) * NUQ + bn];
      bi.x = Ui[((size_t)b * P2 + kb) * NUQ + bn];
    }
    if (kb + 1 < P2) {
      br.y = Ur[((size_t)b * P2 + kb + 1) * NUQ + bn];
      bi.y = Ui[((size_t)b * P2 + kb + 1) * NUQ + bn];
    }
    v2f nai = -ai;
    dr = wmma4(ar, br, dr);
    dr = wmma4(nai, bi, dr);
    di = wmma4(ar, bi, di);
    di = wmma4(ai, br, di);
  }
  for (int r = 0; r < 8; ++r) {
    const int u = m0 + r + half * 8;
    const size_t idx = ((size_t)b * NG + u) * NUQ + bn;
    Vr[idx] = dr[r];
    Vi[idx] = di[r];
  }
}

// ------- Stage E: out[b,u,v,o] = C2[v,:] x Vr[b,u,:,o] + S2[v,:] x Vi -------
// C2/S2 tables staged in LDS.
__global__ __launch_bounds__(256) void k_inv_cols(float* __restrict__ out,
                                                  const float* __restrict__ ws) {
  __shared__ float lC[NG * LPE];
  __shared__ float lS[NG * LPE];

  const float* C2 = ws + OFF_C2;
  const float* S2 = ws + OFF_S2;
  for (int idx = threadIdx.x; idx < NG * KEP; idx += 256) {  // 2560 = 10*256
    const int row = idx / KEP, col = idx % KEP;
    lC[row * LPE + col] = C2[idx];
    lS[row * LPE + col] = S2[idx];
  }
  __syncthreads();

  const int wid  = (int)((blockIdx.x * blockDim.x + threadIdx.x) >> 5);
  const int lane = (int)(threadIdx.x & 31);
  const int half = lane >> 4, l15 = lane & 15;

  const int nt = wid & 1;
  int t = wid >> 1;
  const int mt = t & 7;
  t >>= 3;
  const int u = t & 127;
  const int b = t >> 7;

  const float* Vr = ws + OFF_Y;
  const float* Vi = Vr + VSZ;
  const size_t vbase = ((size_t)b * NG + u) * NUQ;

  const int m0 = mt * 16, n0 = nt * 16;
  const int am = m0 + l15;  // v
  const int bn = n0 + l15;  // o

  v8f d = {};
  for (int k0 = 0; k0 < KEP; k0 += 4) {
    const int kb = k0 + half * 2;
    v2f ac = {lC[am * LPE + kb], lC[am * LPE + kb + 1]};
    v2f as = {lS[am * LPE + kb], lS[am * LPE + kb + 1]};
    v2f br = {}, bi = {};
    if (kb < MODES) {
      br.x = Vr[vbase + (size_t)kb * COUT + bn];
      bi.x = Vi[vbase + (size_t)kb * COUT + bn];
    }
    if (kb + 1 < MODES) {
      br.y = Vr[vbase + (size_t)(kb + 1) * COUT + bn];
      bi.y = Vi[vbase + (size_t)(kb + 1) * COUT + bn];
    }
    d = wmma4(ac, br, d);
    d = wmma4(as, bi, d);
  }
  for (int r = 0; r < 8; ++r) {
    const int v = m0 + r + half * 8;
    out[(((size_t)b * NG + u) * NG + v) * COUT + bn] = d[r];
  }
}

// ---------------- host launcher ----------------
extern "C" void kernel_launch(void* const* d_in, const int* in_sizes, int n_in,
                              void* d_out, int out_size, void* d_ws, size_t ws_size,
                              hipStream_t stream) {
  (void)in_sizes; (void)n_in; (void)out_size;
  const float* x    = (const float*)d_in[0];
  const float* temb = (const float*)d_in[1];
  const float* w1r  = (const float*)d_in[2];
  const float* w1i  = (const float*)d_in[3];
  const float* w2r  = (const float*)d_in[4];
  const float* w2i  = (const float*)d_in[5];
  const float* k1r  = (const float*)d_in[6];
  const float* k1i  = (const float*)d_in[7];
  const float* k2r  = (const float*)d_in[8];
  const float* k2i  = (const float*)d_in[9];
  float* out = (float*)d_out;
  float* ws  = (float*)d_ws;
  if (ws_size < WS_FLOATS * sizeof(float)) return;  // need ~45 MB scratch

  k_twiddle<<<(PM * NG + 255) / 256, 256, 0, stream>>>(ws);
  k_tmod<<<(4 * BATCH * MODES + 255) / 256, 256, 0, stream>>>(temb, k1r, k1i, k2r,
                                                              k2i, ws);
  // 8 warps (tiles) per 256-thread block; tile counts are exact multiples of 8.
  k_fwd_rows<<<(BATCH * 3 * 256) / 8, 256, 0, stream>>>(x, ws);
  k_fwd_cols<<<(BATCH * P2 * 4) / 8, 256, 0, stream>>>(ws);
  k_specmul<<<(P2 * MODES * 4 + 7) / 8, 256, 0, stream>>>(w1r, w1i, w2r, w2i, ws);
  k_inv_rows<<<(BATCH * 8 * 34) / 8, 256, 0, stream>>>(ws);
  k_inv_cols<<<(BATCH * NG * 8 * 2) / 8, 256, 0, stream>>>(out, ws);
}